// FullAttention_38568806318136
// MI455X (gfx1250) — compile-verified
//
#include <hip/hip_runtime.h>
#include <math.h>

typedef __attribute__((ext_vector_type(2))) float v2f;
typedef __attribute__((ext_vector_type(4))) float v4f;
typedef __attribute__((ext_vector_type(8))) float v8f;
typedef __attribute__((ext_vector_type(4))) unsigned int v4u;
typedef __attribute__((ext_vector_type(4))) int v4i;
typedef __attribute__((ext_vector_type(8))) int v8i;

#define Bn 4
#define Ln 2048
#define Sn 2048
#define Hn 8
#define En 64
#define Dn 64
#define TS 16      // WMMA tile edge
#define NW 8       // waves per block
#define NT 256     // threads per block

// Dynamic LDS layout (floats):
//  sS : 16*Sn      score/prob strip           (128 KB)
//  sQ : 16*En      raw Q tile                 (4 KB)   <- TDM target
//  sP : NW*16*Dn   per-wave V partials        (32 KB)
//  sR : NT         reduction scratch
//  sMax,sSum : 16 each
#define LDS_FLOATS (TS*Sn + TS*En + NW*TS*Dn + NT + 32)

__global__ void __launch_bounds__(NT, 1)
fullattn_fwd(const float* __restrict__ Q, const float* __restrict__ K,
             const float* __restrict__ Vv, const float* __restrict__ Mk,
             float* __restrict__ Aout, float* __restrict__ Oout)
{
    extern __shared__ float lds[];
    float* sS   = lds;
    float* sQ   = sS + TS*Sn;
    float* sP   = sQ + TS*En;
    float* sR   = sP + NW*TS*Dn;
    float* sMax = sR + NT;
    float* sSum = sMax + TS;

    const int tid  = threadIdx.x;
    const int lane = tid & 31;
    const int wave = tid >> 5;

    int bid = blockIdx.x;                 // B*H*(L/16) = 4096 blocks
    const int lt = bid & (Ln/TS - 1); bid /= (Ln/TS);
    const int h  = bid & (Hn - 1);    bid /= Hn;
    const int b  = bid;
    const int l0 = lt * TS;

    const float scale  = 0.125f;                       // 1/sqrt(E)
    const float qscale = scale * 0.001f;               // folded in after GEMM1
    const float LNEG   = 23.0258509299404568f * scale; // log(1e10)*scale

    // ---- stage raw Q tile into LDS via the Tensor Data Mover ----
    const float* Qtile = Q + (((size_t)b*Ln + l0)*Hn + h)*En;
#if __has_builtin(__builtin_amdgcn_tensor_load_to_lds) && __has_builtin(__builtin_amdgcn_s_wait_tensorcnt)
    if (wave == 0) {
        const unsigned long long ga = (unsigned long long)(const void*)Qtile;
        // D# group0: count=1 | lds_addr | global_addr[31:0] | global_addr[56:32], type=2
        v4u g0 = { 1u,
                   (unsigned)(TS*Sn*4),                       // byte offset of sQ in LDS
                   (unsigned)(ga & 0xFFFFFFFFull),
                   (unsigned)((ga >> 32) & 0x1FFFFFFull) | (2u << 30) };
        // D# group1: data_size=4B; tensor_dim0=En; tensor_dim1=TS; tile_dim0=En;
        //            tile_dim1=TS; tensor_dim0_stride=Hn*En (elements)
        v8i g1 = { (int)(2u << 16),
                   (int)((unsigned)En << 16),
                   (int)((unsigned)TS << 16),
                   (int)((unsigned)En << 16),
                   (int)TS,
                   (int)(Hn*En),
                   0, 0 };
        v4i z4 = {0,0,0,0};
        v8i z8 = {0,0,0,0,0,0,0,0};
        __builtin_amdgcn_tensor_load_to_lds(g0, g1, z4, z4, z8, 0);
        __builtin_amdgcn_s_wait_tensorcnt(0);
    }
#else
    for (int idx = tid; idx < TS*En; idx += NT) {
        const int r = idx >> 6, e = idx & 63;
        sQ[idx] = Qtile[(size_t)r*Hn*En + e];
    }
#endif
    __syncthreads();

    // Fragment lane mapping for fp32 WMMA (ISA 7.12.2):
    //   lanes 0-15 : row = lane,    K-pair {k, k+1}
    //   lanes16-31 : row = lane-16, K-pair {k+2, k+3}
    const int arow = lane & 15;
    const int aoff = (lane >> 4) << 1;          // 0 or 2
    const int cn   = lane & 15;                 // C-frag: N = lane (mod 16)
    const int cmt  = (lane >> 4) << 3;          // C-frag: M base 0 or 8

    // ================= Pass 1: scores = Q K^T, scale+mask, -> LDS ==============
    for (int st = wave; st < Sn/TS; st += NW) {
        const int s0 = st * TS;
        const float* Krow = K + (((size_t)b*Sn + s0)*Hn + h)*En
                              + (size_t)(lane & 15) * (Hn*En);
        if (st + NW < Sn/TS)
            __builtin_prefetch(Krow + (size_t)NW*TS*Hn*En, 0, 1);

        v8f c = {0.f,0.f,0.f,0.f,0.f,0.f,0.f,0.f};
#pragma unroll
        for (int k = 0; k < En; k += 4) {
            v2f a  = *(const v2f*)&sQ[arow*En + k + aoff];
            v2f bf = *(const v2f*)&Krow[k + aoff];
            c = __builtin_amdgcn_wmma_f32_16x16x4_f32(
                    false, a, false, bf, (short)0, c, false, false);
        }
#pragma unroll
        for (int r = 0; r < 8; r++) {
            const int m = cmt + r;
            const float mk = Mk[((size_t)b*Ln + (l0 + m))*Sn + (s0 + cn)];
            sS[m*Sn + (s0 + cn)] = (c[r]*qscale)*mk + (mk - 1.0f)*LNEG;
        }
    }
    __syncthreads();

    // ================= Pass 2: softmax in LDS ==================================
    {
        const int row = tid >> 4;
        const int cg  = tid & 15;
        float mx = -3.0e38f;
        for (int j = 0; j < Sn/64; j++) {
            v4f v = *(const v4f*)&sS[row*Sn + cg*4 + 64*j];
            mx = fmaxf(mx, fmaxf(fmaxf(v.x, v.y), fmaxf(v.z, v.w)));
        }
        sR[tid] = mx;
        __syncthreads();
        if (tid < TS) {
            float m2 = sR[tid*16];
#pragma unroll
            for (int j = 1; j < 16; j++) m2 = fmaxf(m2, sR[tid*16 + j]);
            sMax[tid] = m2;
        }
        __syncthreads();
        const float rm = sMax[row];
        float sum = 0.f;
        for (int j = 0; j < Sn/64; j++) {
            const int idx = row*Sn + cg*4 + 64*j;
            v4f v = *(const v4f*)&sS[idx];
            v4f e;
            e.x = __expf(v.x - rm); e.y = __expf(v.y - rm);
            e.z = __expf(v.z - rm); e.w = __expf(v.w - rm);
            sum += (e.x + e.y) + (e.z + e.w);
            *(v4f*)&sS[idx] = e;
        }
        sR[tid] = sum;
        __syncthreads();
        if (tid < TS) {
            float s2 = 0.f;
#pragma unroll
            for (int j = 0; j < 16; j++) s2 += sR[tid*16 + j];
            sSum[tid] = 1.0f / s2;
        }
        __syncthreads();
    }
    // normalize in place + wide non-temporal store of A (the critical store)
    for (int q4 = tid; q4 < TS*Sn/4; q4 += NT) {
        const int row = q4 >> 9;               // Sn/4 = 512 vec4 per row
        const float rinv = sSum[row];
        v4f p = *(const v4f*)&sS[q4*4];
        p.x *= rinv; p.y *= rinv; p.z *= rinv; p.w *= rinv;
        *(v4f*)&sS[q4*4] = p;
        v4f* dst = (v4f*)(Aout + (((size_t)b*Hn + h)*Ln + (l0 + row))*Sn
                               + (size_t)(q4 & 511)*4);
        __builtin_nontemporal_store(p, dst);
    }
    __syncthreads();

    // ================= Pass 3: O = P * V, K=Sn split across waves ==============
    v8f acc[4];
#pragma unroll
    for (int nt = 0; nt < 4; nt++) acc[nt] = (v8f){0.f,0.f,0.f,0.f,0.f,0.f,0.f,0.f};

    const int kbeg = wave * (Sn / NW);
    const int kend = kbeg + (Sn / NW);
    for (int k = kbeg; k < kend; k += 4) {
        v2f a = *(const v2f*)&sS[arow*Sn + k + aoff];
        const float* V0 = Vv + (((size_t)b*Sn + (k + aoff))*Hn + h)*Dn;
        const float* V1 = V0 + (size_t)Hn*Dn;
#pragma unroll
        for (int nt = 0; nt < 4; nt++) {
            v2f bf;
            bf.x = V0[nt*16 + (lane & 15)];   // B[k+aoff  ][n]
            bf.y = V1[nt*16 + (lane & 15)];   // B[k+aoff+1][n]
            acc[nt] = __builtin_amdgcn_wmma_f32_16x16x4_f32(
                          false, a, false, bf, (short)0, acc[nt], false, false);
        }
    }
#pragma unroll
    for (int nt = 0; nt < 4; nt++)
#pragma unroll
        for (int r = 0; r < 8; r++)
            sP[(wave*TS + (cmt + r))*Dn + nt*16 + cn] = acc[nt][r];
    __syncthreads();

    // reduce across waves, one v4f per thread, write O
    {
        const int m  = tid >> 4;               // (tid*4) >> 6
        const int d4 = (tid & 15) * 4;
        v4f sum = {0.f, 0.f, 0.f, 0.f};
#pragma unroll
        for (int w = 0; w < NW; w++)
            sum += *(const v4f*)&sP[(w*TS + m)*Dn + d4];
        *(v4f*)(Oout + (((size_t)b*Ln + (l0 + m))*Hn + h)*Dn + d4) = sum;
    }
}

extern "C" void kernel_launch(void* const* d_in, const int* in_sizes, int n_in,
                              void* d_out, int out_size, void* d_ws, size_t ws_size,
                              hipStream_t stream) {
    (void)in_sizes; (void)n_in; (void)d_ws; (void)ws_size; (void)out_size;
    const float* Q  = (const float*)d_in[0];
    const float* K  = (const float*)d_in[1];
    const float* Vv = (const float*)d_in[2];
    const float* Mk = (const float*)d_in[3];
    float* Oout = (float*)d_out;                                   // [B,L,H,D]
    float* Aout = Oout + (size_t)Bn*Ln*Hn*Dn;                      // [B,H,L,S]

    const int blocks = Bn * Hn * (Ln / TS);                        // 4096
    const size_t shmem = (size_t)LDS_FLOATS * sizeof(float);       // ~165 KB
    fullattn_fwd<<<blocks, NT, shmem, stream>>>(Q, K, Vv, Mk, Aout, Oout);
}